// Loss_55490977464946
// MI455X (gfx1250) — compile-verified
//
#include <hip/hip_runtime.h>
#include <hip/hip_bf16.h>
#include <math.h>

typedef float f4  __attribute__((ext_vector_type(4)));
typedef float v2f __attribute__((ext_vector_type(2)));
typedef float v8f __attribute__((ext_vector_type(8)));

#define CL_T 128
#define CL_N 2048
#define CL_D 256
#define TAU_F 0.1f

__device__ __forceinline__ float dot4(f4 a, f4 b) {
  return a.x * b.x + a.y * b.y + a.z * b.z + a.w * b.w;
}

// jax.nn.softplus, numerically stable
__device__ __forceinline__ float softplus(float x) {
  return fmaxf(x, 0.f) + log1pf(expf(-fabsf(x)));
}

// ---------------------------------------------------------------------------
// Kernel 1: BCE partial sums.  -(t*clamp(log p,-100) + (1-t)*clamp(log1p(-p),-100))
// ---------------------------------------------------------------------------
__global__ void bce_partial_kernel(const float* __restrict__ in,
                                   const float* __restrict__ tg,
                                   float* __restrict__ partial, int B) {
  __shared__ float red[256];
  float acc = 0.f;
  for (int i = blockIdx.x * blockDim.x + threadIdx.x; i < B;
       i += gridDim.x * blockDim.x) {
    float p = in[i], t = tg[i];
    float lp   = fmaxf(logf(p), -100.f);
    float l1mp = fmaxf(log1pf(-p), -100.f);
    acc -= t * lp + (1.f - t) * l1mp;
  }
  red[threadIdx.x] = acc;
  __syncthreads();
  for (int s = 128; s > 0; s >>= 1) {
    if ((int)threadIdx.x < s) red[threadIdx.x] += red[threadIdx.x + s];
    __syncthreads();
  }
  if (threadIdx.x == 0) partial[blockIdx.x] = red[0];
}

// ---------------------------------------------------------------------------
// Kernel 2: contrastive-loss streaming kernel.  One wave32 per n.
// Each lane owns 8 of the D=256 features (two contiguous f4 chunks:
// d = 4*lane and d = 128+4*lane -> each b128 load is a fully contiguous
// 512B wave transaction).  Cumsum s lives in registers; per-t we do two
// pairs of cross-lane shuffle reductions (norms, then pos/neg dots).
// Non-temporal loads keep the 512MB single-use stream out of L2, and the
// next t is double-buffered in registers with a prefetch two steps ahead.
// ---------------------------------------------------------------------------
__global__ void cl_partial_kernel(const float* __restrict__ h1,
                                  const float* __restrict__ h2,
                                  float* __restrict__ partial) {
  const int lane = threadIdx.x & 31;
  const int wave = threadIdx.x >> 5;
  const int n = blockIdx.x * 8 + wave;  // blockDim.x == 256 -> 8 waves/block

  const size_t t4 = (size_t)CL_N * CL_D / 4;  // f4 stride between t-slices
  const f4* p1 = (const f4*)(h1 + (size_t)n * CL_D);
  const f4* p2 = (const f4*)(h2 + (size_t)n * CL_D);

  f4 s0 = {0.f, 0.f, 0.f, 0.f};
  f4 s1 = {0.f, 0.f, 0.f, 0.f};
  float acc = 0.f;

  f4 a0 = __builtin_nontemporal_load(p1 + lane);
  f4 a1 = __builtin_nontemporal_load(p1 + 32 + lane);
  f4 c0 = __builtin_nontemporal_load(p2 + lane);
  f4 c1 = __builtin_nontemporal_load(p2 + 32 + lane);

  for (int t = 0; t < CL_T; ++t) {
    f4 na0 = a0, na1 = a1, nc0 = c0, nc1 = c1;
    if (t + 1 < CL_T) {
      const f4* q1 = p1 + (size_t)(t + 1) * t4;
      const f4* q2 = p2 + (size_t)(t + 1) * t4;
      na0 = __builtin_nontemporal_load(q1 + lane);
      na1 = __builtin_nontemporal_load(q1 + 32 + lane);
      nc0 = __builtin_nontemporal_load(q2 + lane);
      nc1 = __builtin_nontemporal_load(q2 + 32 + lane);
      if (t + 2 < CL_T) {
        __builtin_prefetch(p1 + (size_t)(t + 2) * t4 + lane, 0, 1);
        __builtin_prefetch(p2 + (size_t)(t + 2) * t4 + lane, 0, 1);
      }
    }

    // squared norms over D (wave32 reduction)
    float sq1 = dot4(a0, a0) + dot4(a1, a1);
    float sq2 = dot4(c0, c0) + dot4(c1, c1);
#pragma unroll
    for (int off = 16; off > 0; off >>= 1) {
      sq1 += __shfl_xor(sq1, off, 32);
      sq2 += __shfl_xor(sq2, off, 32);
    }
    float inv1 = 1.f / fmaxf(sqrtf(sq1), 1e-12f);
    float inv2 = 1.f / fmaxf(sqrtf(sq2), 1e-12f);

    f4 n0 = a0 * inv1;
    f4 n1 = a1 * inv1;
    s0 += n0;
    s1 += n1;

    float pd = dot4(n0, s0) + dot4(n1, s1);
    float nd = (dot4(c0, s0) + dot4(c1, s1)) * inv2;
#pragma unroll
    for (int off = 16; off > 0; off >>= 1) {
      pd += __shfl_xor(pd, off, 32);
      nd += __shfl_xor(nd, off, 32);
    }

    // s_ref = cumsum/(t+1); pos = dot(n1, s_ref)/TAU  ==  pd / ((t+1)*TAU)
    float scale = 1.f / ((float)(t + 1) * TAU_F);
    acc += softplus(-pd * scale) + softplus(nd * scale);

    a0 = na0; a1 = na1; c0 = nc0; c1 = nc1;
  }

  if (lane == 0) partial[n] = acc;
}

// ---------------------------------------------------------------------------
// Kernel 3: finalize (single wave32).  The 2048 CL partials are reduced with
// a chained V_WMMA_F32_16X16X4_F32 accumulation (B = ones -> each WMMA folds
// 64 f32 partials into the C tile at full precision; sum of all 256 tile
// entries = 16 * grand total).  BCE partials reduced by shuffle.
// ---------------------------------------------------------------------------
__global__ void finalize_kernel(const float* __restrict__ ws,
                                float* __restrict__ out, int B) {
  const int lane = threadIdx.x;  // exactly 32 threads, EXEC all-ones

  // 64 BCE partials
  float b = ws[lane] + ws[lane + 32];
#pragma unroll
  for (int off = 16; off > 0; off >>= 1) b += __shfl_xor(b, off, 32);

  const float* clp = ws + 64;
  float clsum;
#if __has_builtin(__builtin_amdgcn_wmma_f32_16x16x4_f32)
  v8f c = {};
  v2f ones = {1.f, 1.f};
  for (int chunk = 0; chunk < 32; ++chunk) {
    v2f a;
    a.x = clp[chunk * 64 + 2 * lane];
    a.y = clp[chunk * 64 + 2 * lane + 1];
    // D = A x Ones + C : every column of D accumulates the row-sums of A.
    c = __builtin_amdgcn_wmma_f32_16x16x4_f32(false, a, false, ones,
                                              (short)0, c, false, false);
  }
  float s = c[0] + c[1] + c[2] + c[3] + c[4] + c[5] + c[6] + c[7];
#pragma unroll
  for (int off = 16; off > 0; off >>= 1) s += __shfl_xor(s, off, 32);
  clsum = s * (1.f / 16.f);  // 16 identical columns
#else
  float s = 0.f;
  for (int i = lane; i < CL_N; i += 32) s += clp[i];
#pragma unroll
  for (int off = 16; off > 0; off >>= 1) s += __shfl_xor(s, off, 32);
  clsum = s;
#endif

  if (lane == 0)
    out[0] = b / (float)B + clsum / (2.f * (float)CL_N * (float)CL_T);
}

// ---------------------------------------------------------------------------
extern "C" void kernel_launch(void* const* d_in, const int* in_sizes, int n_in,
                              void* d_out, int out_size, void* d_ws,
                              size_t ws_size, hipStream_t stream) {
  const float* input  = (const float*)d_in[0];
  const float* target = (const float*)d_in[1];
  const float* h1     = (const float*)d_in[2];
  const float* h2     = (const float*)d_in[3];
  float* ws  = (float*)d_ws;   // [0..63] BCE partials, [64..64+2047] CL partials
  float* out = (float*)d_out;
  const int B = in_sizes[0];

  bce_partial_kernel<<<64, 256, 0, stream>>>(input, target, ws, B);
  cl_partial_kernel<<<CL_N / 8, 256, 0, stream>>>(h1, h2, ws + 64);
  finalize_kernel<<<1, 32, 0, stream>>>(ws, out, B);
}